// DynamicLightweightConv1dBTC_53171695125228
// MI455X (gfx1250) — compile-verified
//
#include <hip/hip_runtime.h>

// ---------------------------------------------------------------------------
// DynamicLightweightConv1d (fairseq dynamic conv, eval, weight_softmax, causal)
//   B=2, T=2048, C=1024, H=16, K=31, P=30, R=64
// Plan:
//   K1: fp32 WMMA GEMM  w_raw = x @ w_lin^T  (M=4096, N=496, K=1024),
//       fused per-(row,head) softmax over 31 taps -> d_ws (4096 x 496 f32)
//   K2: banded causal conv with LDS tiling of x window + weights
// ---------------------------------------------------------------------------

typedef __attribute__((ext_vector_type(2))) float v2f;
typedef __attribute__((ext_vector_type(8))) float v8f;

#define BATCH   2
#define T_SEQ   2048
#define C_DIM   1024
#define HEADS   16
#define KSIZE   31
#define RDIM    64
#define N_COLS  (HEADS * KSIZE)   // 496
#define N_TILES (N_COLS / 16)     // 31 exactly
#define M_ROWS  (BATCH * T_SEQ)   // 4096

// GEMM tiling
#define KC    128                 // K-chunk staged in LDS
#define KCP   132                 // padded LDS pitch: lane stride 4 banks -> conflict-free b64
#define NPAD  500                 // logits LDS pitch

// ------------------------- Kernel 1: GEMM + softmax ------------------------
// grid.x = M/16 = 256 blocks, 256 threads = 8 waves.
// wave w owns N-tiles {4w..4w+3}; wave 7's 4th tile is clamped to tile 30
// (recomputed redundantly) so the hot loop is 100% branch-free / exec-uniform.
__global__ __launch_bounds__(256) void dynconv_gemm_softmax(
    const float* __restrict__ x,      // (M, C) row-major = (B,T,C)
    const float* __restrict__ w_lin,  // (N, C) row-major
    float* __restrict__ wsm)          // (M, N) softmaxed weights
{
    __shared__ float smem[16 * NPAD]; // 32 KB; first 16*KCP floats reused as A-stage
    float* As = smem;                 // [16][KCP]

    const int tid  = threadIdx.x;
    const int lane = tid & 31;
    const int wave = tid >> 5;
    const int lrow = lane & 15;            // M (for A/C) or N (for B) index
    const int koff = (lane >> 4) << 1;     // 0 | 2 : K sub-offset per ISA layout
    const int m0   = blockIdx.x * 16;

    // Per-tile effective indices (wave 7, i=3 -> duplicate of tile 30)
    int nt0 = wave * 4 + 0;
    int nt1 = wave * 4 + 1;
    int nt2 = wave * 4 + 2;
    int nt3 = wave * 4 + 3; if (nt3 >= N_TILES) nt3 = N_TILES - 1;

    // Hoisted B row pointers: element (k, n) of B == w_lin[n*C + k]
    const float* bp0 = w_lin + (size_t)(nt0 * 16 + lrow) * C_DIM + koff;
    const float* bp1 = w_lin + (size_t)(nt1 * 16 + lrow) * C_DIM + koff;
    const float* bp2 = w_lin + (size_t)(nt2 * 16 + lrow) * C_DIM + koff;
    const float* bp3 = w_lin + (size_t)(nt3 * 16 + lrow) * C_DIM + koff;

    v8f acc0 = (v8f){}, acc1 = (v8f){}, acc2 = (v8f){}, acc3 = (v8f){};

    for (int kc = 0; kc < C_DIM; kc += KC) {
        __syncthreads();
        // Cooperative A stage: 16 rows x 128 floats = 512 float4, 2 per thread
        #pragma unroll
        for (int it = 0; it < 2; ++it) {
            int idx = tid + it * 256;      // float4 index, 32 float4 per row
            int r   = idx >> 5;
            int c4  = (idx & 31) << 2;
            float4 v = *(const float4*)(x + (size_t)(m0 + r) * C_DIM + kc + c4);
            *(float4*)(As + r * KCP + c4) = v;
        }
        // Prefetch next A chunk (16 rows x 8 cachelines) into near cache
        // while this chunk's 32 k-steps run. global_prefetch_b8 path.
        if (kc + KC < C_DIM && tid < 128) {
            int r = tid >> 3;
            int l = tid & 7;
            __builtin_prefetch(x + (size_t)(m0 + r) * C_DIM + kc + KC + l * 16, 0, 0);
        }
        __syncthreads();

        #pragma unroll 4
        for (int ks = 0; ks < KC; ks += 4) {
            // A fragment 16x4: lane<16 -> K{ks,ks+1}, lane>=16 -> K{ks+2,ks+3}
            v2f a  = *(const v2f*)(As + lrow * KCP + ks + koff);
            // Issue all 4 B-fragment loads up front, then 4 WMMAs
            v2f b0 = *(const v2f*)(bp0 + kc + ks);
            v2f b1 = *(const v2f*)(bp1 + kc + ks);
            v2f b2 = *(const v2f*)(bp2 + kc + ks);
            v2f b3 = *(const v2f*)(bp3 + kc + ks);
            acc0 = __builtin_amdgcn_wmma_f32_16x16x4_f32(false, a, false, b0, (short)0, acc0, false, false);
            acc1 = __builtin_amdgcn_wmma_f32_16x16x4_f32(false, a, false, b1, (short)0, acc1, false, false);
            acc2 = __builtin_amdgcn_wmma_f32_16x16x4_f32(false, a, false, b2, (short)0, acc2, false, false);
            acc3 = __builtin_amdgcn_wmma_f32_16x16x4_f32(false, a, false, b3, (short)0, acc3, false, false);
        }
    }

    // ---- dump logits to LDS (C layout: VGPR j -> M = j + 8*(lane>=16), N = lane%16)
    __syncthreads();
    const int rbase = (lane >> 4) << 3;
    {
        int col0 = nt0 * 16 + lrow;
        int col1 = nt1 * 16 + lrow;
        int col2 = nt2 * 16 + lrow;
        #pragma unroll
        for (int j = 0; j < 8; ++j) {
            smem[(rbase + j) * NPAD + col0] = acc0[j];
            smem[(rbase + j) * NPAD + col1] = acc1[j];
            smem[(rbase + j) * NPAD + col2] = acc2[j];
        }
        if (wave * 4 + 3 < N_TILES) {       // skip wave 7's duplicate tile
            int col3 = nt3 * 16 + lrow;
            #pragma unroll
            for (int j = 0; j < 8; ++j)
                smem[(rbase + j) * NPAD + col3] = acc3[j];
        }
    }
    __syncthreads();

    // ---- softmax: 256 groups = 16 rows x 16 heads, one per thread
    {
        const int row = tid >> 4;
        const int h   = tid & 15;
        const float* p = smem + row * NPAD + h * KSIZE;
        float mx = p[0];
        #pragma unroll
        for (int k = 1; k < KSIZE; ++k) mx = fmaxf(mx, p[k]);
        float e[KSIZE];
        float s = 0.f;
        #pragma unroll
        for (int k = 0; k < KSIZE; ++k) { e[k] = expf(p[k] - mx); s += e[k]; }
        float inv = 1.f / s;
        float* o = wsm + (size_t)(m0 + row) * N_COLS + h * KSIZE;
        #pragma unroll
        for (int k = 0; k < KSIZE; ++k) o[k] = e[k] * inv;
    }
}

// ------------------------- Kernel 2: banded causal conv --------------------
// grid = (T/16, H, B), 256 threads. Each block: 16 t's x 64 channels of head h.
#define TBLK  16
#define SROWS (TBLK + KSIZE - 1)  // 46

__global__ __launch_bounds__(256) void dynconv_apply(
    const float* __restrict__ x,    // (B,T,C)
    const float* __restrict__ wsm,  // (M, N) softmaxed weights
    float* __restrict__ out)        // (B,T,C)
{
    __shared__ float xs[SROWS * RDIM];   // 11.5 KB x-window
    __shared__ float wl[TBLK * KSIZE];   // 2 KB weights

    const int tid = threadIdx.x;
    const int t0  = blockIdx.x * TBLK;
    const int h   = blockIdx.y;
    const int b   = blockIdx.z;
    const size_t base = (size_t)b * T_SEQ * C_DIM + (size_t)h * RDIM;

    // Vectorized window stage: 46 rows x 16 float4 = 736 float4 over 256 threads
    for (int idx = tid; idx < SROWS * (RDIM / 4); idx += 256) {
        int sl = idx >> 4;            // row 0..45
        int c4 = (idx & 15) << 2;     // float4 col
        int s  = t0 - (KSIZE - 1) + sl;
        float4 v = make_float4(0.f, 0.f, 0.f, 0.f);
        if (s >= 0)
            v = *(const float4*)(x + base + (size_t)s * C_DIM + c4);
        *(float4*)(xs + sl * RDIM + c4) = v;
    }
    for (int idx = tid; idx < TBLK * KSIZE; idx += 256) {
        int tl = idx / KSIZE;
        int k  = idx % KSIZE;
        wl[idx] = wsm[(size_t)(b * T_SEQ + t0 + tl) * N_COLS + h * KSIZE + k];
    }
    __syncthreads();

    const int r  = tid & 63;
    const int tq = tid >> 6;   // 0..3
    for (int tt = 0; tt < 4; ++tt) {
        int tl = tq * 4 + tt;  // 0..15
        float a = 0.f;
        #pragma unroll
        for (int k = 0; k < KSIZE; ++k)
            a = fmaf(wl[tl * KSIZE + k], xs[(tl + k) * RDIM + r], a);
        out[base + (size_t)(t0 + tl) * C_DIM + r] = a;
    }
}

// --------------------------------- launch ----------------------------------
extern "C" void kernel_launch(void* const* d_in, const int* in_sizes, int n_in,
                              void* d_out, int out_size, void* d_ws, size_t ws_size,
                              hipStream_t stream) {
    (void)in_sizes; (void)n_in; (void)out_size; (void)ws_size;
    const float* x     = (const float*)d_in[0];
    const float* w_lin = (const float*)d_in[1];
    float* out = (float*)d_out;
    float* wsm = (float*)d_ws;   // needs 4096*496*4 = 8.1 MB scratch

    dynconv_gemm_softmax<<<dim3(M_ROWS / 16), 256, 0, stream>>>(x, w_lin, wsm);
    dynconv_apply<<<dim3(T_SEQ / TBLK, HEADS, BATCH), 256, 0, stream>>>(x, wsm, out);
}